// SparseAttentionPattern_64278480552149
// MI455X (gfx1250) — compile-verified
//
#include <hip/hip_runtime.h>
#include <math.h>

// SparseAttentionPattern mask generator for MI455X (gfx1250).
//
// Output: N x N float32 (1.0 / 0.0), N = 16384 -> 1.07 GB of stores, ~0 bytes
// read. Store-bandwidth bound: floor = 1.07e9 B / 23.3e12 B/s ~= 46 us.
// Key optimization this round: wave-uniform chunk classification. A wave32
// chunk spans 128 consecutive k, and nearly every chunk of a row is entirely
// all-ones (window/global), all-zeros (past causal bound), or dilated-only
// (precomputed per-thread pattern). Classify with readfirstlane -> SGPR
// compares (scalar pipe) so the hot paths issue ~2 VALU + 1 NT b128 store
// per 512 B, keeping the kernel pinned to HBM write bandwidth instead of
// VALU issue rate.

#define WINDOW_HALF 2048   // WINDOW_SIZE / 2
#define DILATION_MASK 63   // DILATION_STRIDE - 1 (power of two)
#define GLOBAL_TOKENS 128

typedef float float4v __attribute__((ext_vector_type(4)));

__global__ __launch_bounds__(256) void sparse_mask_kernel(float* __restrict__ out,
                                                          int N) {
  const int q = blockIdx.x;          // one block per query row
  if (q >= N) return;
  const int lo = q - WINDOW_HALF;    // window lower bound under causality
  float* __restrict__ row = out + (size_t)q * (size_t)N;

  // k4 = tid*4 + i*1024, so (k4 & 63) is loop-invariant: within a float4
  // chunk only element 0 can be a dilation column, and only if tid % 16 == 0.
  const bool d0 = ((threadIdx.x & 15) == 0);
  const float4v vOnes = {1.0f, 1.0f, 1.0f, 1.0f};
  const float4v vZero = {0.0f, 0.0f, 0.0f, 0.0f};
  const float4v vDil  = {d0 ? 1.0f : 0.0f, 0.0f, 0.0f, 0.0f};

  for (int k4 = threadIdx.x * 4; k4 + 3 < N; k4 += 256 * 4) {
    // Wave-uniform chunk bounds: lane0 of each wave has k4 = multiple of 128.
    const int u0   = __builtin_amdgcn_readfirstlane(k4);
    const int umax = u0 + 127;
    float4v v;
    if (umax <= q && (u0 >= lo || umax < GLOBAL_TOKENS)) {
      // Fully causal AND (fully inside window OR fully inside global region).
      v = vOnes;
    } else if (u0 > q) {
      // Entire chunk beyond the causal bound.
      v = vZero;
    } else if (umax < lo && u0 >= GLOBAL_TOKENS) {
      // Pre-window causal zone (umax < lo <= q implies causal): only the
      // dilation columns survive -> loop-invariant per-thread pattern.
      v = vDil;
    } else {
      // Boundary chunk (at most 2 waves per row): full per-element predicate.
#pragma unroll
      for (int j = 0; j < 4; ++j) {
        const int k = k4 + j;
        const bool t = (k <= q) &
                       ((k >= lo) | ((k & DILATION_MASK) == 0) | (k < GLOBAL_TOKENS));
        v[j] = t ? 1.0f : 0.0f;
      }
    }
    __builtin_nontemporal_store(v, (float4v*)(row + k4));
  }

  // Tail for N not divisible by 4 (not hit for N = 16384, kept for generality).
  const int ntail = N & 3;
  if (ntail) {
    const int base = N & ~3;
    for (int j = threadIdx.x; j < ntail; j += 256) {
      const int k = base + j;
      const bool t = (k <= q) &
                     ((k >= lo) | ((k & DILATION_MASK) == 0) | (k < GLOBAL_TOKENS));
      __builtin_nontemporal_store(t ? 1.0f : 0.0f, row + k);
    }
  }
}

extern "C" void kernel_launch(void* const* d_in, const int* in_sizes, int n_in,
                              void* d_out, int out_size, void* d_ws, size_t ws_size,
                              hipStream_t stream) {
  (void)d_in; (void)in_sizes; (void)n_in; (void)d_ws; (void)ws_size;
  // seq_len lives in device memory (d_in[0]); during graph capture we cannot
  // do a synchronous copy, but out_size = N*N lets us recover N on the host.
  const int N = (int)(sqrt((double)out_size) + 0.5);
  float* out = (float*)d_out;

  dim3 grid(N);     // one block per row
  dim3 block(256);  // 8 wave32s
  sparse_mask_kernel<<<grid, block, 0, stream>>>(out, N);
}